// LongRangeProj_75703093559323
// MI455X (gfx1250) — compile-verified
//
#include <hip/hip_runtime.h>
#include <math.h>

typedef __attribute__((ext_vector_type(2))) float v2f;
typedef __attribute__((ext_vector_type(8))) float v8f;

// Problem constants from the reference
#define BB 2
#define CC 64
#define HH 64
#define WW 64
// NH=NW=8, STRIDE=8 -> 64 source nodes per (b,c)

// acos via A&S 4.4.46: acos(x) = sqrt(1-x)*poly(x) on [0,1], |err| <= 2e-8.
// Branch-free, raw v_sqrt (argument in [0,1], well conditioned).
__device__ __forceinline__ float fast_acos(float x)
{
    const float ax = __builtin_fabsf(x);
    const float s  = __builtin_amdgcn_sqrtf(1.0f - ax);
    float p;
    p = fmaf(ax, -0.0012624911f, 0.0066700901f);
    p = fmaf(ax, p, -0.0170881256f);
    p = fmaf(ax, p,  0.0308918810f);
    p = fmaf(ax, p, -0.0501743046f);
    p = fmaf(ax, p,  0.0889789874f);
    p = fmaf(ax, p, -0.2145988016f);
    p = fmaf(ax, p,  1.5707963050f);
    const float r = s * p;
    return (x < 0.0f) ? (3.14159274101257f - r) : r;
}

__global__ __launch_bounds__(256) void
LongRangeProj_kernel(const float* __restrict__ x,
                     const float* __restrict__ radius_mean,
                     const float* __restrict__ angle_mean,
                     const float* __restrict__ radius_std,
                     const float* __restrict__ angle_std,
                     float* __restrict__ out)
{
    __shared__ float  s_cs[64];
    __shared__ float  s_sn[64];
    __shared__ float  s_d [64];
    __shared__ float2 s_rx[64];   // (|radius_mean|, x) packed -> one ds_load_b64

    const int bc  = blockIdx.x;      // 0..127 == b*64 + c
    const int c   = bc & 63;
    const int tid = threadIdx.x;

    // Per-channel Gaussian scales (fold both exps into one exponent)
    const float rs    = radius_std[c];
    const float as    = angle_std[c];
    const float inv2r = 1.0f / (2.0f * (rs * rs + 0.01f));
    const float inv2a = 1.0f / (2.0f * (as * as + 1e-4f));

    // Phase 1: stage per-node params in LDS (64 nodes per (b,c))
    if (tid < 64) {
        const int idx = bc * 64 + tid;
        const float am = angle_mean[idx];
        const float cs = cosf(am);
        const float sn = sinf(am);
        const float ox = (float)((tid & 7) << 3);   // 8*j
        const float oy = (float)((tid >> 3) << 3);  // 8*i
        s_cs[tid] = cs;
        s_sn[tid] = sn;
        s_d [tid] = -(cs * ox + sn * oy);           // num = cs*w + sn*h + d
        s_rx[tid] = make_float2(fabsf(radius_mean[idx]), x[idx]);
    }
    __syncthreads();

    const int lane = tid & 31;
    const int wave = tid >> 5;          // 0..7
    const int hi   = lane >> 4;         // 0: lanes 0-15, 1: lanes 16-31
    const int lmod = lane & 15;

    // A operands for the 4 ij-tiles (16x4 f32 A layout):
    // lanes 0-15 hold K=0 (VGPR0) / K=1 (VGPR1); lanes 16-31 hold K=2 / K=3.
    // Row M = ij: (cos, sin, d, 0)
    v2f aop[4];
#pragma unroll
    for (int t = 0; t < 4; ++t) {
        const int m = t * 16 + lmod;
        aop[t][0] = hi ? s_d[m] : s_cs[m];
        aop[t][1] = hi ? 0.0f   : s_sn[m];
    }

    const v8f czero = {};

    // Each wave covers 8 rows (h) x 4 tiles of 16 columns
    for (int t = 0; t < 32; ++t) {
        const int h     = wave * 8 + (t >> 2);
        const int wBase = (t & 3) << 4;

        // B operand (4x16): column N = lmod is (w, h, 1, 0) in K slots
        // matching A's layout (lo lanes K=0/1, hi lanes K=2/3).
        v2f bop;
        bop[0] = hi ? 1.0f : (float)(wBase + lmod);
        bop[1] = hi ? 0.0f : (float)h;

        const float wpx = (float)(wBase + lmod); // pixel column this lane owns
        const float hpx = (float)h;

        float best = -3.0e38f;

#pragma unroll
        for (int it = 0; it < 4; ++it) {
            // num[M=r+8*hi][N=lmod] for ij-tile `it`
            v8f num = __builtin_amdgcn_wmma_f32_16x16x4_f32(
                false, aop[it], false, bop, (short)0, czero, false, false);

#pragma unroll
            for (int r = 0; r < 8; ++r) {
                const int    ij = it * 16 + r + hi * 8;
                const float2 rx = s_rx[ij];
                const float  ox = (float)((ij & 7) << 3);
                const float  oy = (float)((ij >> 3) << 3);

                const float fx  = wpx - ox;
                const float fy  = hpx - oy;
                const float fn2 = fmaf(fx, fx, fy * fy);

                // One v_rsq gives both 1/fn and fn (fn = fn2 * rsqrt(fn2)).
                // At the node center (fn2==0) both would be NaN; the two
                // selects (same compare) force cosang=1, fn=0 branch-free.
                const float rrt = __builtin_amdgcn_rsqf(fn2);
                const bool  ctr = (fn2 == 0.0f);
                float fn     = ctr ? 0.0f : fn2 * rrt;
                float cosang = ctr ? 1.0f : num[r] * rrt;
                cosang = fminf(1.0f, fmaxf(-1.0f, cosang));

                const float ang = fast_acos(cosang);
                const float dr  = fn - rx.x;
                const float e   = -fmaf(dr * dr, inv2r, ang * ang * inv2a);
                const float wgt = rx.y * __expf(e);
                best = fmaxf(best, wgt);
            }
        }

        // Combine the two M-halves (lane L and L^16 hold same pixel column)
        const float other = __shfl_xor(best, 16, 32);
        best = fmaxf(best, other);

        if (lane < 16) {
            out[(bc * 64 + h) * 64 + (wBase + lmod)] = best;
        }
    }
}

extern "C" void kernel_launch(void* const* d_in, const int* in_sizes, int n_in,
                              void* d_out, int out_size, void* d_ws, size_t ws_size,
                              hipStream_t stream)
{
    (void)in_sizes; (void)n_in; (void)out_size; (void)d_ws; (void)ws_size;
    const float* x           = (const float*)d_in[0];
    const float* radius_mean = (const float*)d_in[1];
    const float* angle_mean  = (const float*)d_in[2];
    const float* radius_std  = (const float*)d_in[3];
    const float* angle_std   = (const float*)d_in[4];
    float* out = (float*)d_out;

    LongRangeProj_kernel<<<BB * CC, 256, 0, stream>>>(
        x, radius_mean, angle_mean, radius_std, angle_std, out);
}